// grnn_54958401520104
// MI455X (gfx1250) — compile-verified
//
#include <hip/hip_runtime.h>

// ---------------------------------------------------------------------------
// GRNN-style fused op on MI455X (gfx1250), full fp32 via V_WMMA_F32_16X16X4_F32.
//
//   Ax  = input[B,2048] @ A[2048,49]                  (GEMM 1)
//   out = (Ax[:,:,None]*state[B,49,64]) x G[49,64,64] (gated GEMM 2, K=3136)
//
// Roofline: ~20 GFLOP / ~690 MB -> ~29 FLOP/B, right at the f32-WMMA balance
// point of the chip, so f32 WMMA (not bf16) is the correct precision/perf pick.
// ---------------------------------------------------------------------------

typedef float v2f __attribute__((ext_vector_type(2)));
typedef float v8f __attribute__((ext_vector_type(8)));

#define B_ROWS   32768
#define IN_F     2048
#define NUM_A    49
#define HID      64
#define KTOT     (NUM_A * HID)   // 3136
#define NPAD     64              // N padded to 64 (4 x 16 WMMA col-tiles)
#define WAVES_PB 8               // 256 threads = 8 wave32

__device__ __forceinline__ v8f wmma_f32_k4(v2f a, v2f b, v8f c) {
  // D(16x16,f32) += A(16x4,f32) x B(4x16,f32)
  return __builtin_amdgcn_wmma_f32_16x16x4_f32(
      /*neg_a=*/false, a, /*neg_b=*/false, b,
      /*c_mod=*/(short)0, c, /*reuse_a=*/false, /*reuse_b=*/false);
}

// --- prep: A[2048,49] -> ATpad[64,2048] (zero-padded, n-major) --------------
__global__ __launch_bounds__(256) void grnn_prep_at(const float* __restrict__ A,
                                                    float* __restrict__ ATpad) {
  int idx = blockIdx.x * 256 + threadIdx.x;
  if (idx >= NPAD * IN_F) return;
  int n = idx / IN_F;
  int k = idx % IN_F;
  ATpad[idx] = (n < NUM_A) ? A[(size_t)k * NUM_A + n] : 0.0f;
}

// --- prep: G[3136,64] -> Gt[64,3136] (n-major) ------------------------------
__global__ __launch_bounds__(256) void grnn_prep_gt(const float* __restrict__ G,
                                                    float* __restrict__ Gt) {
  int idx = blockIdx.x * 256 + threadIdx.x;
  if (idx >= NPAD * KTOT) return;
  int n = idx / KTOT;
  int k = idx % KTOT;
  Gt[idx] = G[(size_t)k * NPAD + n];
}

// --- GEMM 1: AxPad[B,64] = input[B,2048] @ ATpad^T --------------------------
// One wave owns a 16x64 tile; K loop 2048/4 = 512 steps, 4 WMMA per step.
__global__ __launch_bounds__(256) void grnn_ax_kernel(const float* __restrict__ input,
                                                      const float* __restrict__ ATpad,
                                                      float* __restrict__ AxPad) {
  const int lane = threadIdx.x & 31;
  const int wave = threadIdx.x >> 5;
  const int half = lane >> 4;    // lane group: K pair {0,1} vs {2,3}
  const int ln   = lane & 15;    // M row (A-frag) / N col (B-frag)
  const int koff = half * 2;
  const int row0 = (blockIdx.x * WAVES_PB + wave) * 16;

  const float* arow = input + (size_t)(row0 + ln) * IN_F;
  const float* b0r  = ATpad + (size_t)(0 * 16 + ln) * IN_F;
  const float* b1r  = ATpad + (size_t)(1 * 16 + ln) * IN_F;
  const float* b2r  = ATpad + (size_t)(2 * 16 + ln) * IN_F;
  const float* b3r  = ATpad + (size_t)(3 * 16 + ln) * IN_F;

  v8f acc0 = {}, acc1 = {}, acc2 = {}, acc3 = {};

#pragma unroll 4
  for (int k = 0; k < IN_F; k += 4) {
    v2f af = *(const v2f*)(arow + k + koff);
    v2f b0 = *(const v2f*)(b0r + k + koff);
    v2f b1 = *(const v2f*)(b1r + k + koff);
    v2f b2 = *(const v2f*)(b2r + k + koff);
    v2f b3 = *(const v2f*)(b3r + k + koff);
    acc0 = wmma_f32_k4(af, b0, acc0);
    acc1 = wmma_f32_k4(af, b1, acc1);
    acc2 = wmma_f32_k4(af, b2, acc2);
    acc3 = wmma_f32_k4(af, b3, acc3);
  }

  // C/D layout: VGPR v, lanes0-15 -> M=v, lanes16-31 -> M=v+8, N=lane&15
#pragma unroll
  for (int v = 0; v < 8; ++v) {
    float* p = AxPad + (size_t)(row0 + v + half * 8) * NPAD + ln;
    p[0]  = acc0[v];
    p[16] = acc1[v];
    p[32] = acc2[v];
    p[48] = acc3[v];
  }
}

// --- GEMM 2 (fused gate): out[B,64] = (Ax*state) @ Gt^T ---------------------
// K = 3136 iterated a-major (49 chunks of 64); the gate scalar Ax[row,a] is
// folded into the A-fragment so `state` streams through HBM exactly once.
__global__ __launch_bounds__(256) void grnn_out_kernel(const float* __restrict__ state,
                                                       const float* __restrict__ Gt,
                                                       const float* __restrict__ AxPad,
                                                       float* __restrict__ out) {
  const int lane = threadIdx.x & 31;
  const int wave = threadIdx.x >> 5;
  const int half = lane >> 4;
  const int ln   = lane & 15;
  const int koff = half * 2;
  const int row0 = (blockIdx.x * WAVES_PB + wave) * 16;

  const float* srow  = state + (size_t)(row0 + ln) * KTOT;
  const float* axrow = AxPad + (size_t)(row0 + ln) * NPAD;
  const float* g0r   = Gt + (size_t)(0 * 16 + ln) * KTOT;
  const float* g1r   = Gt + (size_t)(1 * 16 + ln) * KTOT;
  const float* g2r   = Gt + (size_t)(2 * 16 + ln) * KTOT;
  const float* g3r   = Gt + (size_t)(3 * 16 + ln) * KTOT;

  v8f acc0 = {}, acc1 = {}, acc2 = {}, acc3 = {};

  for (int a = 0; a < NUM_A; ++a) {
    const float ax = axrow[a];
    const int   kb = a * HID;
    // prefetch next a-chunk of this row's state (emits global_prefetch_b8)
    __builtin_prefetch(srow + kb + HID, 0, 1);
#pragma unroll 4
    for (int h = 0; h < HID; h += 4) {
      v2f sf = *(const v2f*)(srow + kb + h + koff);
      v2f af = ax * sf;  // gate folded into A-fragment
      v2f b0 = *(const v2f*)(g0r + kb + h + koff);
      v2f b1 = *(const v2f*)(g1r + kb + h + koff);
      v2f b2 = *(const v2f*)(g2r + kb + h + koff);
      v2f b3 = *(const v2f*)(g3r + kb + h + koff);
      acc0 = wmma_f32_k4(af, b0, acc0);
      acc1 = wmma_f32_k4(af, b1, acc1);
      acc2 = wmma_f32_k4(af, b2, acc2);
      acc3 = wmma_f32_k4(af, b3, acc3);
    }
  }

#pragma unroll
  for (int v = 0; v < 8; ++v) {
    float* p = out + (size_t)(row0 + v + half * 8) * HID + ln;
    p[0]  = acc0[v];
    p[16] = acc1[v];
    p[32] = acc2[v];
    p[48] = acc3[v];
  }
}

// ---------------------------------------------------------------------------
extern "C" void kernel_launch(void* const* d_in, const int* in_sizes, int n_in,
                              void* d_out, int out_size, void* d_ws, size_t ws_size,
                              hipStream_t stream) {
  const float* input = (const float*)d_in[0];  // [32768, 2048]
  const float* state = (const float*)d_in[1];  // [32768, 49, 64]
  const float* A     = (const float*)d_in[2];  // [2048, 49]
  const float* G     = (const float*)d_in[3];  // [49, 64, 64]
  float* out = (float*)d_out;                  // [32768, 64]

  // workspace layout (256B aligned): ATpad 512KB | Gt 802KB | AxPad 8MB
  char* ws = (char*)d_ws;
  const size_t at_bytes = ((size_t)NPAD * IN_F * sizeof(float) + 255) & ~(size_t)255;
  const size_t gt_bytes = ((size_t)NPAD * KTOT * sizeof(float) + 255) & ~(size_t)255;
  float* ATpad = (float*)(ws);
  float* Gt    = (float*)(ws + at_bytes);
  float* AxPad = (float*)(ws + at_bytes + gt_bytes);

  {
    int n = NPAD * IN_F;
    grnn_prep_at<<<(n + 255) / 256, 256, 0, stream>>>(A, ATpad);
  }
  {
    int n = NPAD * KTOT;
    grnn_prep_gt<<<(n + 255) / 256, 256, 0, stream>>>(G, Gt);
  }

  const int blocks = B_ROWS / (16 * WAVES_PB);  // 256 blocks, 8 waves each
  grnn_ax_kernel<<<blocks, 256, 0, stream>>>(input, ATpad, AxPad);
  grnn_out_kernel<<<blocks, 256, 0, stream>>>(state, Gt, AxPad, out);
}